// DijkstraGridSolver_45320494907667
// MI455X (gfx1250) — compile-verified
//
#include <hip/hip_runtime.h>

#define AS1 __attribute__((address_space(1)))
#define AS3 __attribute__((address_space(3)))

#ifndef __has_builtin
#define __has_builtin(x) 0
#endif

typedef __attribute__((ext_vector_type(4))) int v4i;

constexpr int N   = 32;
constexpr int NN  = N * N;      // 1024 cells per batch element
constexpr int T   = 256;        // threads per block = 8 wave32
constexpr int CPT = NN / T;     // 4 cells per thread (strided by T)

__device__ __forceinline__ unsigned long long shfl_xor_u64(unsigned long long v, int mask) {
  unsigned lo = (unsigned)(v & 0xFFFFFFFFull);
  unsigned hi = (unsigned)(v >> 32);
  lo = __shfl_xor(lo, mask, 32);
  hi = __shfl_xor(hi, mask, 32);
  return ((unsigned long long)hi << 32) | (unsigned long long)lo;
}

__global__ __launch_bounds__(T) void dijkstra_grid_kernel(
    const float* __restrict__ W, float* __restrict__ Out) {
  const int b = blockIdx.x;
  const int t = threadIdx.x;
  const float* wb = W + (size_t)b * NN;

  __shared__ float              wlds[NN];      // staged weights (4 KB)
  __shared__ unsigned long long part[T / 32];  // per-wave argmin partials
  __shared__ unsigned           predl[NN];     // flat predecessor indices
  __shared__ float              pathl[NN];     // output path mask

  // ---- Stage weight tile global -> LDS via CDNA5 async copy (ASYNCcnt) ----
  {
    const float* gp = wb + t * 4;              // 16 B per lane, 4 KB per block
    float*       lp = wlds + t * 4;
#if __has_builtin(__builtin_amdgcn_global_load_async_to_lds_b128) && \
    __has_builtin(__builtin_amdgcn_s_wait_asynccnt)
    __builtin_amdgcn_global_load_async_to_lds_b128(
        (AS1 v4i*)(void*)gp, (AS3 v4i*)(void*)lp, /*offset=*/0, /*cpol=*/0);
    __builtin_amdgcn_s_wait_asynccnt(0);
#else
    unsigned ldsOff = (unsigned)(unsigned long long)(void*)lp;
    asm volatile("global_load_async_to_lds_b128 %0, %1, off"
                 :: "v"(ldsOff), "v"(gp) : "memory");
    asm volatile("s_wait_asynccnt 0" ::: "memory");
#endif
  }
  __syncthreads();

  // ---- Per-thread register state: cells t, t+256, t+512, t+768 ----
  float    dist[CPT];
  float    wreg[CPT];
  unsigned pred[CPT];
  bool     vis[CPT];
#pragma unroll
  for (int i = 0; i < CPT; ++i) {
    const int cell = t + i * T;
    wreg[i] = wlds[cell];
    dist[i] = (cell == 0) ? 0.0f : __builtin_inff();
    pred[i] = 0u;                 // reference: pred initialized to (0,0)
    vis[i]  = false;
  }

  const int wave = t >> 5;
  const int lane = t & 31;

  // ---- n*n Dijkstra pops (exactly mirrors the reference loop) ----
#pragma unroll 1
  for (int iter = 0; iter < NN; ++iter) {
    // local min over owned cells; key = (dist_bits << 32) | flat_idx
    unsigned long long key = ~0ull;
#pragma unroll
    for (int i = 0; i < CPT; ++i) {
      const int   cell = t + i * T;
      const float d    = vis[i] ? __builtin_inff() : dist[i];
      const unsigned long long k =
          ((unsigned long long)__float_as_uint(d) << 32) | (unsigned)cell;
      key = (k < key) ? k : key;
    }
    // wave32 min-reduce (lowest index wins ties, matching jnp.argmin)
#pragma unroll
    for (int off = 16; off >= 1; off >>= 1) {
      const unsigned long long o = shfl_xor_u64(key, off);
      key = (o < key) ? o : key;
    }
    if (lane == 0) part[wave] = key;
    __syncthreads();
    unsigned long long best = part[0];
#pragma unroll
    for (int wv = 1; wv < T / 32; ++wv) {
      const unsigned long long o = part[wv];
      best = (o < best) ? o : best;
    }
    __syncthreads();  // part[] reused next iteration

    const unsigned bidx = (unsigned)(best & 0xFFFFFFFFull);
    const float    cur  = __uint_as_float((unsigned)(best >> 32));
    const int      cy   = (int)(bidx >> 5);
    const int      cx   = (int)(bidx & 31u);

    // mark popped cell visited; relax the 8 neighbors (owner-updates, dup-safe)
#pragma unroll
    for (int i = 0; i < CPT; ++i) {
      const int cell = t + i * T;
      if ((unsigned)cell == bidx) vis[i] = true;
      const int y  = cell >> 5, x = cell & 31;
      const int dy = y - cy,    dx = x - cx;
      const int ady = (dy < 0) ? -dy : dy;
      const int adx = (dx < 0) ? -dx : dx;
      const bool nb = (ady <= 1) && (adx <= 1) && ((ady | adx) != 0);
      if (nb && !vis[i]) {
        const float nd = cur + wreg[i];        // cost of entering neighbor cell
        if (nd < dist[i]) { dist[i] = nd; pred[i] = bidx; }
      }
    }
  }

  // ---- Backtrack (31,31) -> (0,0) through predecessor chain ----
#pragma unroll
  for (int i = 0; i < CPT; ++i) {
    predl[t + i * T] = pred[i];
    pathl[t + i * T] = 0.0f;
  }
  __syncthreads();
  if (t == 0) {
    unsigned c = NN - 1;
    pathl[c] = 1.0f;
#pragma unroll 1
    for (int s = 0; s < NN; ++s) {
      c = predl[c];
      pathl[c] = 1.0f;
      if (c == 0u) break;        // pred[(0,0)] == (0,0): fixed point
    }
  }
  __syncthreads();

  float* ob = Out + (size_t)b * NN;
#pragma unroll
  for (int i = 0; i < CPT; ++i) ob[t + i * T] = pathl[t + i * T];
}

extern "C" void kernel_launch(void* const* d_in, const int* in_sizes, int n_in,
                              void* d_out, int out_size, void* d_ws, size_t ws_size,
                              hipStream_t stream) {
  (void)n_in; (void)out_size; (void)d_ws; (void)ws_size;
  const float* W   = (const float*)d_in[0];
  float*       Out = (float*)d_out;
  const int    b   = in_sizes[0] / NN;   // 128 batch elements
  dijkstra_grid_kernel<<<dim3(b), dim3(T), 0, stream>>>(W, Out);
}